// Network_35871566856832
// MI455X (gfx1250) — compile-verified
//
#include <hip/hip_runtime.h>
#include <math.h>

#define PN 1200
#define HN 250
#define LNODES 150
#define NN 1600
#define TT 52
#define FEATD 64

typedef __attribute__((ext_vector_type(2))) float v2f;
typedef __attribute__((ext_vector_type(8))) float v8f;

__constant__ float c_LOW[18] = {0,0,0,0,0,0,0,0.9f,0.9f,0.9f,0,0,0,0,0,0,0,0};
__constant__ float c_UP[18]  = {10,10,10,10,2,0.02f,0.1f,1,1,1,
                                0.02f,0.02f,0.02f,0.02f,0.002f,0.02f,0.02f,0.01f};

__device__ __forceinline__ float sigmoidf_(float x) { return 1.f / (1.f + __expf(-x)); }

// -------------------------------------------------------------------------
// WMMA GEMM: C[M x Nout] = A[M x K] @ W^T  (W stored row-major Nout x K),
// + bias, optional BN+ReLU epilogue. One wave (32 threads) per 16x16 C tile.
// Uses V_WMMA_F32_16X16X4_F32 (full fp32, wave32).
// A fragment (16x4): lane&15 = M row; lane>>4 selects K pair {0,1}/{2,3}.
// B fragment (4x16): B[k][n] = W[n0+n][k] -> identical per-lane addressing.
// C/D (16x16): VGPR j -> row m0+j (lanes 0-15) / m0+j+8 (lanes 16-31).
// -------------------------------------------------------------------------
__global__ void gemm_wmma_f32(const float* __restrict__ A, int lda,
                              const float* __restrict__ W, int ldw,
                              const float* __restrict__ bias,
                              const float* __restrict__ g, const float* __restrict__ be,
                              const float* __restrict__ mean, const float* __restrict__ var,
                              float* __restrict__ C, int ldc,
                              int K, int bnrelu)
{
    const int m0   = blockIdx.x * 16;
    const int n0   = blockIdx.y * 16;
    const int lane = threadIdx.x;      // 0..31, wave32
    const int half = lane >> 4;        // 0 or 1
    const int l15  = lane & 15;

    v8f acc = {};
    const float* ap = A + (size_t)(m0 + l15) * lda + half * 2;
    const float* wp = W + (size_t)(n0 + l15) * ldw + half * 2;
    for (int k = 0; k < K; k += 4) {
        v2f a = *reinterpret_cast<const v2f*>(ap + k);
        v2f b = *reinterpret_cast<const v2f*>(wp + k);
        // (neg_a, A, neg_b, B, c_mod, C, reuse_a, reuse_b)
        acc = __builtin_amdgcn_wmma_f32_16x16x4_f32(false, a, false, b,
                                                    (short)0, acc, false, false);
    }

    const int col = n0 + l15;
    const float bv = bias[col];
    float gg = 0.f, bb = 0.f, mm = 0.f, inv = 0.f;
    if (bnrelu) {
        gg = g[col]; bb = be[col]; mm = mean[col];
        inv = rsqrtf(var[col] + 1e-5f);
    }
#pragma unroll
    for (int j = 0; j < 8; ++j) {
        const int row = m0 + j + half * 8;
        float x = acc[j] + bv;
        if (bnrelu) x = fmaxf(gg * (x - mm) * inv + bb, 0.f);
        C[(size_t)row * ldc + col] = x;
    }
}

// fc5 (16 -> 1) + sigmoid + clip: one thread per row.
__global__ void fc5_sigmoid_kernel(const float* __restrict__ h4,
                                   const float* __restrict__ w5,
                                   const float* __restrict__ b5,
                                   float* __restrict__ prob)
{
    int i = blockIdx.x * blockDim.x + threadIdx.x;
    if (i >= PN) return;
    float s = b5[0];
#pragma unroll
    for (int k = 0; k < 16; ++k) s += h4[i * 16 + k] * w5[k];
    float p = sigmoidf_(s);
    prob[i] = fminf(fmaxf(p, 0.f), 1.f);
}

// param = LOW + (UP-LOW)*sigmoid(para)
__global__ void param_kernel(const float* __restrict__ para, float* __restrict__ param_out)
{
    int i = threadIdx.x;
    if (i < 18) param_out[i] = c_LOW[i] + (c_UP[i] - c_LOW[i]) * sigmoidf_(para[i]);
}

// t=0: states0 = imp_index[0]*scale; loads0 = states0*imp_index[0]*(mu+sigma*noise)
__global__ void init_state_kernel(const float* __restrict__ imp_index,
                                  const float* __restrict__ noise_std,
                                  const float* __restrict__ prob,
                                  const float* __restrict__ param,
                                  float* __restrict__ s0, float* __restrict__ l0,
                                  float* __restrict__ out_states)
{
    int i = blockIdx.x * blockDim.x + threadIdx.x;
    if (i >= NN) return;
    float scale = (i < PN) ? prob[i] : 1.f;
    float imp0  = imp_index[i];               // row t=0
    float s = imp0 * scale;
    float l = s * imp0 * (param[0] + param[1] * noise_std[i]);
    s0[i] = s;
    l0[i] = l;
    if (i < PN) out_states[i] = s;
}

// One step of the scan. One 256-thread block per output row i (bandwidth-bound):
//   s_new[i] elementwise, then l_new[i] = (As[t,i,:]*tau + diag phi) . l_old + alpha*s_new[i]
// The As stream (532 MB total across the scan, touched once) is loaded with the
// non-temporal hint so it does not evict the hot l_old/prob/param/state vectors
// from WGP$/L2 (192 MB L2 cannot hold it anyway).
__global__ void step_kernel(const float* __restrict__ As,
                            const float* __restrict__ imp_index,
                            const float* __restrict__ prob,
                            const float* __restrict__ param,
                            const float* __restrict__ s_old, const float* __restrict__ l_old,
                            float* __restrict__ s_new, float* __restrict__ l_new,
                            float* __restrict__ out_states, int t)
{
    __shared__ float red[256];
    const int i   = blockIdx.x;
    const int tid = threadIdx.x;

    const float alpha = param[4], beta = param[5], delta = param[6];
    const int ti = (i < PN) ? 0 : ((i < PN + HN) ? 1 : 2);
    const float phi = param[7 + ti];
    float tau0, tau1, tau2;                       // tau[ti][0..2] by source type
    if (ti == 0)      { tau0 = param[10]; tau1 = param[13]; tau2 = param[16]; }
    else if (ti == 1) { tau0 = param[11]; tau1 = param[14]; tau2 = param[17]; }
    else              { tau0 = param[12]; tau1 = param[15]; tau2 = 0.f; }

    // elementwise state update for node i (redundant across threads; cheap)
    const float s   = s_old[i];
    const float l   = l_old[i];
    const float impv = __builtin_nontemporal_load(&imp_index[(size_t)t * NN + i]) *
                       ((i < PN) ? prob[i] : 1.f);
    float s_upd = (1.f - delta) * s + beta * l * (1.f - s) + impv;
    s_upd = fminf(fmaxf(s_upd, 0.f), 1.f);
    const float sn = (i < PN) ? s_upd : fminf(fmaxf(s, 0.f), 1.f);

    // row dot product (coalesced non-temporal stream of the As row)
    const float* Arow = As + ((size_t)t * NN + i) * (size_t)NN;
    float part = 0.f;
    for (int j = tid; j < NN; j += 256) {
        const float a  = __builtin_nontemporal_load(&Arow[j]);
        const int   tj = (j < PN) ? 0 : ((j < PN + HN) ? 1 : 2);
        const float tf = (tj == 0) ? tau0 : ((tj == 1) ? tau1 : tau2);
        float r = a * tf;
        if (j == i) r += phi;
        part += r * l_old[j];
    }
    red[tid] = part;
    __syncthreads();
    for (int off = 128; off > 0; off >>= 1) {
        if (tid < off) red[tid] += red[tid + off];
        __syncthreads();
    }
    if (tid == 0) {
        l_new[i] = red[0] + alpha * sn;
        s_new[i] = sn;
        if (i < PN) out_states[(size_t)t * PN + i] = sn;
    }
}

extern "C" void kernel_launch(void* const* d_in, const int* in_sizes, int n_in,
                              void* d_out_v, int out_size, void* d_ws, size_t ws_size,
                              hipStream_t stream)
{
    const float* features  = (const float*)d_in[0];
    const float* fc1_w     = (const float*)d_in[1];
    const float* fc1_b     = (const float*)d_in[2];
    const float* fc2_w     = (const float*)d_in[3];
    const float* fc2_b     = (const float*)d_in[4];
    const float* bn2_g     = (const float*)d_in[5];
    const float* bn2_b     = (const float*)d_in[6];
    const float* bn2_m     = (const float*)d_in[7];
    const float* bn2_v     = (const float*)d_in[8];
    const float* fc3_w     = (const float*)d_in[9];
    const float* fc3_b     = (const float*)d_in[10];
    const float* bn3_g     = (const float*)d_in[11];
    const float* bn3_b     = (const float*)d_in[12];
    const float* bn3_m     = (const float*)d_in[13];
    const float* bn3_v     = (const float*)d_in[14];
    const float* fc4_w     = (const float*)d_in[15];
    const float* fc4_b     = (const float*)d_in[16];
    const float* fc5_w     = (const float*)d_in[17];
    const float* fc5_b     = (const float*)d_in[18];
    const float* para      = (const float*)d_in[19];
    const float* As        = (const float*)d_in[20];
    const float* imp_index = (const float*)d_in[21];
    const float* noise_std = (const float*)d_in[22];

    float* out        = (float*)d_out_v;
    float* out_states = out;                  // 52*1200
    float* out_prob   = out + TT * PN;        // 1200
    float* out_param  = out + TT * PN + PN;   // 18

    float* ws    = (float*)d_ws;
    float* sbuf0 = ws;                        // N
    float* sbuf1 = ws + NN;                   // N
    float* lbuf0 = ws + 2 * NN;               // N
    float* lbuf1 = ws + 3 * NN;               // N
    float* h1    = ws + 4 * NN;               // 1200*128
    float* h2    = h1 + PN * 128;             // 1200*128
    float* h3    = h2 + PN * 128;             // 1200*64
    float* h4    = h3 + PN * 64;              // 1200*16

    // 1) parameter squash (also final output component)
    param_kernel<<<1, 32, 0, stream>>>(para, out_param);

    // 2) MLP via fp32 WMMA GEMMs (one wave per 16x16 tile)
    gemm_wmma_f32<<<dim3(PN / 16, 128 / 16), 32, 0, stream>>>(
        features, FEATD, fc1_w, FEATD, fc1_b,
        nullptr, nullptr, nullptr, nullptr, h1, 128, FEATD, 0);
    gemm_wmma_f32<<<dim3(PN / 16, 128 / 16), 32, 0, stream>>>(
        h1, 128, fc2_w, 128, fc2_b,
        bn2_g, bn2_b, bn2_m, bn2_v, h2, 128, 128, 1);
    gemm_wmma_f32<<<dim3(PN / 16, 64 / 16), 32, 0, stream>>>(
        h2, 128, fc3_w, 128, fc3_b,
        bn3_g, bn3_b, bn3_m, bn3_v, h3, 64, 128, 1);
    gemm_wmma_f32<<<dim3(PN / 16, 16 / 16), 32, 0, stream>>>(
        h3, 64, fc4_w, 64, fc4_b,
        nullptr, nullptr, nullptr, nullptr, h4, 16, 64, 0);
    fc5_sigmoid_kernel<<<(PN + 255) / 256, 256, 0, stream>>>(h4, fc5_w, fc5_b, out_prob);

    // 3) initial state (t=0)
    init_state_kernel<<<(NN + 255) / 256, 256, 0, stream>>>(
        imp_index, noise_std, out_prob, out_param, sbuf0, lbuf0, out_states);

    // 4) sequential scan: 51 bandwidth-bound fused update+matvec steps
    float* scur = sbuf0; float* lcur = lbuf0;
    float* snxt = sbuf1; float* lnxt = lbuf1;
    for (int t = 1; t < TT; ++t) {
        step_kernel<<<NN, 256, 0, stream>>>(As, imp_index, out_prob, out_param,
                                            scur, lcur, snxt, lnxt, out_states, t);
        float* tmp;
        tmp = scur; scur = snxt; snxt = tmp;
        tmp = lcur; lcur = lnxt; lnxt = tmp;
    }

    (void)in_sizes; (void)n_in; (void)out_size; (void)ws_size;
}